// FrameSimilarity_37400575214077
// MI455X (gfx1250) — compile-verified
//
#include <hip/hip_runtime.h>
#include <hip/hip_bf16.h>

// Problem constants (from reference)
#define B_   4
#define N_   256
#define T_   4096
#define HW_  48          // H*W = 6*8
#define D_   128         // SIM_DIM
#define LK_  101         // LOOKUP
#define PAD_ 50          // (LOOKUP-1)/2
#define OUT_ 128         // OUT_DIM

typedef __attribute__((ext_vector_type(2))) float v2f;
typedef __attribute__((ext_vector_type(8))) float v8f;

// ---------------------------------------------------------------------------
// Kernel 1: spatial mean-pool  x[B,N,T,H,W] -> feat[B,T,N]
// Bandwidth-bound stage (805 MB read). Each thread reads its own 192B
// (three 64B lines, 12x float4) and the block writes 1KB fully coalesced.
// ---------------------------------------------------------------------------
__global__ __launch_bounds__(256) void pool_kernel(const float* __restrict__ x,
                                                   float* __restrict__ feat) {
  const int bt = blockIdx.x;            // over B*T
  const int b  = bt / T_;
  const int t  = bt % T_;
  const int n  = threadIdx.x;           // 0..255
  const size_t base = ((size_t)(b * N_ + n) * T_ + t) * HW_;
  const float4* p = (const float4*)(x + base);
  float s = 0.f;
#pragma unroll
  for (int i = 0; i < HW_ / 4; ++i) {
    float4 v = p[i];
    s += v.x + v.y + v.z + v.w;
  }
  feat[((size_t)b * T_ + t) * N_ + n] = s * (1.0f / (float)HW_);
}

// ---------------------------------------------------------------------------
// Kernel 2: proj = feat @ w_proj^T, then L2-normalize rows.
// One wave32 computes a 16(M) x 128(N) tile via V_WMMA_F32_16X16X4_F32.
//   A 16x4 f32 : lane(m=lane&15), half=lane>>4 holds K = 2*half + {0,1}
//   B 4x16 f32 : lane holds N = ct*16 + (lane&15), half holds K = 2*half+{0,1}
//   C 16x16    : VGPR v, lanes 0-15 -> row v, lanes 16-31 -> row v+8
// ---------------------------------------------------------------------------
__global__ __launch_bounds__(32) void proj_kernel(const float* __restrict__ feat,
                                                  const float* __restrict__ w_proj,
                                                  float* __restrict__ proj) {
  const int m0    = blockIdx.x * 16;    // row tile over B*T
  const int lane  = threadIdx.x;
  const int laneM = lane & 15;
  const int half  = lane >> 4;

  v8f acc[8];
#pragma unroll
  for (int i = 0; i < 8; ++i)
#pragma unroll
    for (int j = 0; j < 8; ++j) acc[i][j] = 0.f;

  const float* arow = feat + (size_t)(m0 + laneM) * N_;
#pragma unroll 4
  for (int k = 0; k < N_; k += 4) {
    v2f a;
    a.x = arow[k + 2 * half + 0];
    a.y = arow[k + 2 * half + 1];
#pragma unroll
    for (int ct = 0; ct < 8; ++ct) {
      const float* brow = w_proj + (size_t)(ct * 16 + laneM) * N_;  // w_proj[d][k]
      v2f bb;
      bb.x = brow[k + 2 * half + 0];
      bb.y = brow[k + 2 * half + 1];
      acc[ct] = __builtin_amdgcn_wmma_f32_16x16x4_f32(
          false, a, false, bb, (short)0, acc[ct], false, false);
    }
  }

  // Row-wise L2 normalization entirely in registers. For C-fragment VGPR v,
  // lanes 0-15 hold row v (cols = lane), lanes 16-31 hold row v+8: xor-shuffle
  // reductions with masks 1/2/4/8 stay within each 16-lane half.
#pragma unroll
  for (int v = 0; v < 8; ++v) {
    float ss = 0.f;
#pragma unroll
    for (int ct = 0; ct < 8; ++ct) { float e = acc[ct][v]; ss += e * e; }
    ss += __shfl_xor(ss, 1, 32);
    ss += __shfl_xor(ss, 2, 32);
    ss += __shfl_xor(ss, 4, 32);
    ss += __shfl_xor(ss, 8, 32);
    const float inv = 1.0f / fmaxf(sqrtf(ss), 1e-12f);
#pragma unroll
    for (int ct = 0; ct < 8; ++ct) acc[ct][v] *= inv;
  }

  // Store proj[B*T, 128]
#pragma unroll
  for (int v = 0; v < 8; ++v) {
    const int row = m0 + v + 8 * half;
#pragma unroll
    for (int ct = 0; ct < 8; ++ct)
      proj[(size_t)row * D_ + ct * 16 + laneM] = acc[ct][v];
  }
}

// ---------------------------------------------------------------------------
// Kernel 3: banded cosine sims + FC + ReLU, one wave per (b, 16-row t-tile).
// Stage A: S[16x128] = P[t0..t0+15] @ Ctx^T, Ctx rows s = t0-50+c (zero rows
//          outside [0,T) reproduce the reference's zero band padding).
// Stage B: band[m][j] = S[m][m+j] (diagonal gather through LDS), then
//          out = relu(band @ w_fc^T + b_fc) with bias preloaded into C.
// ---------------------------------------------------------------------------
__global__ __launch_bounds__(32) void band_fc_kernel(const float* __restrict__ proj,
                                                     const float* __restrict__ w_fc,
                                                     const float* __restrict__ b_fc,
                                                     float* __restrict__ out) {
  __shared__ float s_lds[16 * 132];     // 16 rows, stride 132 (bank-conflict pad)

  const int tile  = blockIdx.x;         // over B * (T/16)
  const int b     = tile / (T_ / 16);
  const int t0    = (tile % (T_ / 16)) * 16;
  const int lane  = threadIdx.x;
  const int laneM = lane & 15;
  const int half  = lane >> 4;
  const float* projB = proj + (size_t)b * T_ * D_;

  // ---- Stage A: similarity strip ----
  v8f acc[8];
#pragma unroll
  for (int i = 0; i < 8; ++i)
#pragma unroll
    for (int j = 0; j < 8; ++j) acc[i][j] = 0.f;

  const float* arow = projB + (size_t)(t0 + laneM) * D_;
#pragma unroll 2
  for (int k = 0; k < D_; k += 4) {
    v2f a;
    a.x = arow[k + 2 * half + 0];
    a.y = arow[k + 2 * half + 1];
#pragma unroll
    for (int ct = 0; ct < 8; ++ct) {
      const int s   = t0 - PAD_ + ct * 16 + laneM;   // ctx row (col index of S)
      const bool ok = (s >= 0) && (s < T_);
      const float* brow = projB + (size_t)(ok ? s : 0) * D_;
      v2f bb;
      bb.x = ok ? brow[k + 2 * half + 0] : 0.f;
      bb.y = ok ? brow[k + 2 * half + 1] : 0.f;
      acc[ct] = __builtin_amdgcn_wmma_f32_16x16x4_f32(
          false, a, false, bb, (short)0, acc[ct], false, false);
    }
  }

  // Spill S to LDS (row-major, padded stride)
#pragma unroll
  for (int v = 0; v < 8; ++v)
#pragma unroll
    for (int ct = 0; ct < 8; ++ct)
      s_lds[(v + 8 * half) * 132 + ct * 16 + laneM] = acc[ct][v];
  __syncthreads();   // single-wave WG: S_NOP, but keeps the compiler honest

  // ---- Stage B: FC over the 101-wide band ----
  v8f d[8];
#pragma unroll
  for (int ct = 0; ct < 8; ++ct) {
    const float bias = b_fc[ct * 16 + laneM];        // bias depends on col only
#pragma unroll
    for (int j = 0; j < 8; ++j) d[ct][j] = bias;
  }

#pragma unroll 2
  for (int k = 0; k < 104; k += 4) {                 // K = 101 padded to 104
    const int m  = laneM;
    const int j0 = k + 2 * half;
    v2f a;
    a.x = (j0     < LK_) ? s_lds[m * 132 + m + j0    ] : 0.f;
    a.y = (j0 + 1 < LK_) ? s_lds[m * 132 + m + j0 + 1] : 0.f;
#pragma unroll
    for (int ct = 0; ct < 8; ++ct) {
      const float* wrow = w_fc + (size_t)(ct * 16 + laneM) * LK_;  // w_fc[o][j]
      v2f bb;
      bb.x = (j0     < LK_) ? wrow[j0    ] : 0.f;
      bb.y = (j0 + 1 < LK_) ? wrow[j0 + 1] : 0.f;
      d[ct] = __builtin_amdgcn_wmma_f32_16x16x4_f32(
          false, a, false, bb, (short)0, d[ct], false, false);
    }
  }

  // ReLU + store out[B, T, 128]
#pragma unroll
  for (int v = 0; v < 8; ++v) {
    const int t = t0 + v + 8 * half;
#pragma unroll
    for (int ct = 0; ct < 8; ++ct)
      out[((size_t)b * T_ + t) * OUT_ + ct * 16 + laneM] = fmaxf(d[ct][v], 0.f);
  }
}

// ---------------------------------------------------------------------------
extern "C" void kernel_launch(void* const* d_in, const int* in_sizes, int n_in,
                              void* d_out, int out_size, void* d_ws, size_t ws_size,
                              hipStream_t stream) {
  const float* x      = (const float*)d_in[0];  // [B,N,T,H,W]
  const float* w_proj = (const float*)d_in[1];  // [D, N]
  const float* w_fc   = (const float*)d_in[2];  // [OUT, LK]
  const float* b_fc   = (const float*)d_in[3];  // [OUT]
  float* out = (float*)d_out;                   // [B, T, OUT]

  float* feat = (float*)d_ws;                   // [B,T,N]   16.78 MB
  float* proj = feat + (size_t)B_ * T_ * N_;    // [B*T, D]   8.39 MB

  pool_kernel<<<B_ * T_, 256, 0, stream>>>(x, feat);
  proj_kernel<<<(B_ * T_) / 16, 32, 0, stream>>>(feat, w_proj, proj);
  band_fc_kernel<<<B_ * (T_ / 16), 32, 0, stream>>>(proj, w_fc, b_fc, out);
}